// DuelingGCN_69140383531493
// MI455X (gfx1250) — compile-verified
//
#include <hip/hip_runtime.h>

// ---------------- problem sizes ----------------
#define NN      50000
#define NE      150000
#define NODE_F  4
#define EDGE_F  11
#define KPAD    12          // K=11 padded to 12 for 3x WMMA f32 16x16x4
#define HID     512
#define HID_OUT 1024
#define VS      2048
#define ASZ     2048
#define N_ACT   1000

typedef __attribute__((ext_vector_type(2))) float v2f;
typedef __attribute__((ext_vector_type(8))) float v8f;

// ---------------- kernels ----------------

__global__ void k_zero(float* p, int n) {
    int i = blockIdx.x * blockDim.x + threadIdx.x;
    if (i < n) p[i] = 0.0f;
}

// degree histograms over real edges (self loops added analytically later)
__global__ void k_degrees(const int* __restrict__ src, const int* __restrict__ dst,
                          float* degO, float* degI) {
    int e = blockIdx.x * blockDim.x + threadIdx.x;
    if (e >= NE) return;
    atomicAdd(&degO[src[e]], 1.0f);
    atomicAdd(&degI[dst[e]], 1.0f);
}

// per-node: deg -> rsqrt, w self-loop init, h0s = nf * rout, agg1 init = self loop
__global__ void k_node_init(const float* __restrict__ nf,
                            float* degO /*->rout*/, float* degI /*->rin*/,
                            float* w, float* h0s, float* agg1) {
    int v = blockIdx.x * blockDim.x + threadIdx.x;
    if (v >= NN) return;
    float ro = rsqrtf(degO[v] + 1.0f);   // +1 self loop
    float ri = rsqrtf(degI[v] + 1.0f);
    degO[v] = ro;
    degI[v] = ri;
    w[v] = ri * (1.0f / (float)NN);      // self-loop contribution to layer-3 weight
#pragma unroll
    for (int k = 0; k < NODE_F; ++k) {
        float f = nf[v * NODE_F + k] * ro;
        h0s[v * NODE_F + k] = f;
        agg1[v * NODE_F + k] = f;        // self-loop term of layer-1 aggregation
    }
}

// layer-1 edge scatter + layer-3 scalar weight scatter
__global__ void k_scatter1(const int* __restrict__ src, const int* __restrict__ dst,
                           const float* __restrict__ h0s, const float* __restrict__ rin,
                           float* agg1, float* w) {
    int e = blockIdx.x * blockDim.x + threadIdx.x;
    if (e >= NE) return;
    int s = src[e], d = dst[e];
#pragma unroll
    for (int k = 0; k < NODE_F; ++k)
        atomicAdd(&agg1[d * NODE_F + k], h0s[s * NODE_F + k]);
    atomicAdd(&w[s], rin[d] * (1.0f / (float)NN));
}

// h1s[v] = rout[v] * relu( (agg1[v] @ W1) * rin[v] + b1 )
__global__ void k_h1(const float* __restrict__ agg1,
                     const float* __restrict__ W1, const float* __restrict__ b1,
                     const float* __restrict__ rin, const float* __restrict__ rout,
                     float* __restrict__ h1s) {
    int v = blockIdx.x * blockDim.x + threadIdx.x;
    if (v >= NN) return;
    float a0 = agg1[v * 4 + 0], a1 = agg1[v * 4 + 1];
    float a2 = agg1[v * 4 + 2], a3 = agg1[v * 4 + 3];
    float ri = rin[v], ro = rout[v];
#pragma unroll
    for (int j = 0; j < EDGE_F; ++j) {
        float t = a0 * W1[0 * EDGE_F + j] + a1 * W1[1 * EDGE_F + j]
                + a2 * W1[2 * EDGE_F + j] + a3 * W1[3 * EDGE_F + j];
        float h = fmaxf(t * ri + b1[j], 0.0f);
        h1s[v * EDGE_F + j] = h * ro;
    }
}

// layer-2 edge scatter with edge features; agg2 padded to stride KPAD (row 11 stays 0)
__global__ void k_scatter2(const int* __restrict__ src, const int* __restrict__ dst,
                           const float* __restrict__ h1s, const float* __restrict__ ef,
                           float* agg2) {
    int e = blockIdx.x * blockDim.x + threadIdx.x;
    if (e >= NE) return;
    int s = src[e], d = dst[e];
#pragma unroll
    for (int j = 0; j < EDGE_F; ++j)
        atomicAdd(&agg2[d * KPAD + j], h1s[s * EDGE_F + j] * ef[e * EDGE_F + j]);
}

// Fused layer-2 GEMM + layer-3 weighted readout:
//   s[0:512] += sum_v  w[v]*rout[v] * relu( (agg2[v] @ W2) * rin[v] + b2 )
// V_WMMA_F32_16X16X4_F32 on 16-node tiles; K padded to 12 (branch-free fragments).
// W2 staged zero-padded [12 x 512] in LDS via gfx1250 async memory->LDS DMA.
__global__ void __launch_bounds__(256)
k_layer2_wmma(const float* __restrict__ agg2, const float* __restrict__ W2,
              const float* __restrict__ b2, const float* __restrict__ rin,
              const float* __restrict__ rout, const float* __restrict__ w,
              float* __restrict__ s_out) {
    __shared__ float lds_w2[KPAD * HID];   // 24 KB, row 11 zeroed
    __shared__ float lds_s[HID];
    __shared__ float lds_b2[HID];

    const int tid = threadIdx.x;

    // async DMA W2 rows 0..10 straight into LDS (ASYNCcnt-tracked, no VGPR bounce)
    for (int i = tid; i < EDGE_F * HID; i += blockDim.x) {
        unsigned lva = (unsigned)(uintptr_t)&lds_w2[i];  // low 32 bits of flat addr = LDS offset
        const float* gsrc = &W2[i];
        asm volatile("global_load_async_to_lds_b32 %0, %1, off"
                     :: "v"(lva), "v"(gsrc) : "memory");
    }
    for (int i = tid; i < HID; i += blockDim.x) {
        lds_w2[EDGE_F * HID + i] = 0.0f;                 // pad row K=11
        lds_s[i] = 0.0f;
        lds_b2[i] = b2[i];
    }
    asm volatile("s_wait_asynccnt 0x0" ::: "memory");
    __syncthreads();

    const int lane   = tid & 31;
    const int wvblk  = tid >> 5;
    const int nwblk  = blockDim.x >> 5;
    const int gwave  = blockIdx.x * nwblk + wvblk;
    const int nwaves = gridDim.x * nwblk;
    const int ntiles = NN / 16;              // 3125 exactly

    const int mrow = lane & 15;              // A-frag: row this lane carries
    const int sel  = lane >> 4;              // A/B-frag: K sub-pair selector (0/1)
    const int mb   = sel * 8;                // D-frag: row base for this lane half

    for (int t = gwave; t < ntiles; t += nwaves) {
        const int base = t * 16;

        // per-node scale factors for the D-fragment rows this lane touches
        float rinv[8], coefv[8];
#pragma unroll
        for (int r = 0; r < 8; ++r) {
            int v = base + mb + r;
            rinv[r]  = rin[v];
            coefv[r] = w[v] * rout[v];
        }

        // A fragments: padded agg2 tile [16 x 12], unconditional 8B loads
        const v2f* arp = (const v2f*)(agg2 + (size_t)(base + mrow) * KPAD);
        v2f afrag[3];
#pragma unroll
        for (int c = 0; c < 3; ++c)
            afrag[c] = arp[2 * c + sel];

        // sweep 32 column blocks of the LDS-resident padded W2
        for (int j = 0; j < HID / 16; ++j) {
            const int col = (j << 4) + mrow;
            v8f acc = {};
#pragma unroll
            for (int c = 0; c < 3; ++c) {
                int k0 = 4 * c + 2 * sel;
                v2f bv;
                bv.x = lds_w2[k0 * HID + col];
                bv.y = lds_w2[(k0 + 1) * HID + col];
                acc = __builtin_amdgcn_wmma_f32_16x16x4_f32(
                          false, afrag[c], false, bv, (short)0, acc, false, false);
            }
            // epilogue: rin-scale + bias + relu + (w*rout) weight, reduce rows
            float local = 0.0f;
            float bias = lds_b2[col];
#pragma unroll
            for (int r = 0; r < 8; ++r) {
                float e = fmaxf(acc[r] * rinv[r] + bias, 0.0f);
                local += e * coefv[r];
            }
            atomicAdd(&lds_s[col], local);
        }
    }

    __syncthreads();
    for (int i = tid; i < HID; i += blockDim.x)
        atomicAdd(&s_out[i], lds_s[i]);
}

// g_emb = s @ W3 + b3   [1,512]@[512,1024]
__global__ void k_gemb(const float* __restrict__ s, const float* __restrict__ W3,
                       const float* __restrict__ b3, float* __restrict__ gemb) {
    __shared__ float ls[HID];
    for (int i = threadIdx.x; i < HID; i += blockDim.x) ls[i] = s[i];
    __syncthreads();
    int j = blockIdx.x * blockDim.x + threadIdx.x;
    if (j >= HID_OUT) return;
    float t = b3[j];
    for (int k = 0; k < HID; ++k) t += ls[k] * W3[k * HID_OUT + j];
    gemb[j] = t;
}

// hv = relu(g @ Wv1 + bv1), ha = relu(g @ Wa1 + ba1)
__global__ void k_heads1(const float* __restrict__ g,
                         const float* __restrict__ Wv1, const float* __restrict__ bv1,
                         const float* __restrict__ Wa1, const float* __restrict__ ba1,
                         float* __restrict__ hv, float* __restrict__ ha) {
    __shared__ float lg[HID_OUT];
    for (int i = threadIdx.x; i < HID_OUT; i += blockDim.x) lg[i] = g[i];
    __syncthreads();
    int j = blockIdx.x * blockDim.x + threadIdx.x;
    if (j < VS) {
        float t = bv1[j];
        for (int k = 0; k < HID_OUT; ++k) t += lg[k] * Wv1[k * VS + j];
        hv[j] = fmaxf(t, 0.0f);
    } else if (j < VS + ASZ) {
        int jj = j - VS;
        float t = ba1[jj];
        for (int k = 0; k < HID_OUT; ++k) t += lg[k] * Wa1[k * ASZ + jj];
        ha[jj] = fmaxf(t, 0.0f);
    }
}

// v = hv . Wv2 + bv2 (single-block reduction)
__global__ void k_value(const float* __restrict__ hv, const float* __restrict__ Wv2,
                        const float* __restrict__ bv2, float* __restrict__ vout) {
    __shared__ float red[256];
    float p = 0.0f;
    for (int k = threadIdx.x; k < VS; k += 256) p += hv[k] * Wv2[k];
    red[threadIdx.x] = p;
    __syncthreads();
    for (int sft = 128; sft > 0; sft >>= 1) {
        if (threadIdx.x < sft) red[threadIdx.x] += red[threadIdx.x + sft];
        __syncthreads();
    }
    if (threadIdx.x == 0) vout[0] = red[0] + bv2[0];
}

// a[j] = ha @ Wa2[:,j] + ba2[j];  asum += a[j]
__global__ void k_heads2(const float* __restrict__ ha, const float* __restrict__ Wa2,
                         const float* __restrict__ ba2, float* __restrict__ a,
                         float* __restrict__ asum) {
    __shared__ float lh[ASZ];
    for (int i = threadIdx.x; i < ASZ; i += blockDim.x) lh[i] = ha[i];
    __syncthreads();
    int j = blockIdx.x * blockDim.x + threadIdx.x;
    if (j >= N_ACT) return;
    float t = ba2[j];
    for (int k = 0; k < ASZ; ++k) t += lh[k] * Wa2[k * N_ACT + j];
    a[j] = t;
    atomicAdd(asum, t);
}

// out = v + a - mean(a)
__global__ void k_final(const float* __restrict__ a, const float* __restrict__ v,
                        const float* __restrict__ asum, float* __restrict__ out) {
    int j = blockIdx.x * blockDim.x + threadIdx.x;
    if (j >= N_ACT) return;
    out[j] = v[0] + a[j] - asum[0] * (1.0f / (float)N_ACT);
}

// ---------------- launch ----------------

extern "C" void kernel_launch(void* const* d_in, const int* in_sizes, int n_in,
                              void* d_out, int out_size, void* d_ws, size_t ws_size,
                              hipStream_t stream) {
    (void)in_sizes; (void)n_in; (void)out_size; (void)ws_size;

    const float* nf  = (const float*)d_in[0];
    const float* ef  = (const float*)d_in[1];
    const float* W1  = (const float*)d_in[2];
    const float* b1  = (const float*)d_in[3];
    const float* W2  = (const float*)d_in[4];
    const float* b2  = (const float*)d_in[5];
    const float* W3  = (const float*)d_in[6];
    const float* b3  = (const float*)d_in[7];
    const float* Wv1 = (const float*)d_in[8];
    const float* bv1 = (const float*)d_in[9];
    const float* Wv2 = (const float*)d_in[10];
    const float* bv2 = (const float*)d_in[11];
    const float* Wa1 = (const float*)d_in[12];
    const float* ba1 = (const float*)d_in[13];
    const float* Wa2 = (const float*)d_in[14];
    const float* ba2 = (const float*)d_in[15];
    const int*   src = (const int*)d_in[16];
    const int*   dst = (const int*)d_in[17];
    float* out = (float*)d_out;

    // workspace layout (floats); zero-region is the leading block
    float* ws   = (float*)d_ws;
    float* degO = ws;                       // N    -> rout after k_node_init
    float* degI = ws + NN;                  // N    -> rin  after k_node_init
    float* agg2 = ws + 2 * NN;              // 12N  (padded stride)
    float* svec = ws + 14 * NN;             // 512
    float* asum = ws + 14 * NN + 512;       // 1
    float* vval = ws + 14 * NN + 513;       // 1
    const int nzero = 14 * NN + 514;
    float* wgt  = ws + nzero;               // N
    float* h0s  = wgt + NN;                 // 4N
    float* agg1 = h0s + 4 * NN;             // 4N
    float* h1s  = agg1 + 4 * NN;            // 11N
    float* gemb = h1s + 11 * NN;            // 1024
    float* hv   = gemb + HID_OUT;           // 2048
    float* ha   = hv + VS;                  // 2048
    float* avec = ha + ASZ;                 // 1000

    const int T = 256;
    k_zero<<<(nzero + T - 1) / T, T, 0, stream>>>(ws, nzero);
    k_degrees<<<(NE + T - 1) / T, T, 0, stream>>>(src, dst, degO, degI);
    k_node_init<<<(NN + T - 1) / T, T, 0, stream>>>(nf, degO, degI, wgt, h0s, agg1);
    k_scatter1<<<(NE + T - 1) / T, T, 0, stream>>>(src, dst, h0s, degI, agg1, wgt);
    k_h1<<<(NN + T - 1) / T, T, 0, stream>>>(agg1, W1, b1, degI, degO, h1s);
    k_scatter2<<<(NE + T - 1) / T, T, 0, stream>>>(src, dst, h1s, ef, agg2);
    k_layer2_wmma<<<128, 256, 0, stream>>>(agg2, W2, b2, degI, degO, wgt, svec);
    k_gemb<<<(HID_OUT + T - 1) / T, T, 0, stream>>>(svec, W3, b3, gemb);
    k_heads1<<<(VS + ASZ + T - 1) / T, T, 0, stream>>>(gemb, Wv1, bv1, Wa1, ba1, hv, ha);
    k_value<<<1, 256, 0, stream>>>(hv, Wv2, bv2, vval);
    k_heads2<<<(N_ACT + T - 1) / T, T, 0, stream>>>(ha, Wa2, ba2, avec, asum);
    k_final<<<(N_ACT + T - 1) / T, T, 0, stream>>>(avec, vval, asum, out);
}